// distmult_head_18245021074000
// MI455X (gfx1250) — compile-verified
//
#include <hip/hip_runtime.h>
#include <hip/hip_bf16.h>
#include <stdint.h>

// H is fixed at 256 by the reference.
#define H 256

typedef __attribute__((ext_vector_type(2))) float v2f;
typedef __attribute__((ext_vector_type(8))) float v8f;

// ---------------------------------------------------------------------------
// K0: zero the workspace accumulators (counts[nsrc], left_sum[H], v[H]).
// ---------------------------------------------------------------------------
__global__ void k_zero(unsigned int* __restrict__ p, int n) {
    int i = blockIdx.x * blockDim.x + threadIdx.x;
    if (i < n) p[i] = 0u;
}

// ---------------------------------------------------------------------------
// K1: histogram of source indices. counts[src_idx[e]] += 1
// 300K scattered u32 atomics over a 200KB array -> L2-resident, cheap.
// ---------------------------------------------------------------------------
__global__ void k_hist(const int* __restrict__ src_idx, unsigned int* __restrict__ counts, int E) {
    int i = blockIdx.x * blockDim.x + threadIdx.x;
    if (i < E) atomicAdd(&counts[src_idx[i]], 1u);
}

// ---------------------------------------------------------------------------
// K2: left_sum[h] = sum_n counts[n] * x_src[n][h]
// blockDim = H(256): thread h owns column h; rows are grid-strided so global
// reads are fully coalesced (256 consecutive floats per row per block).
// One f32 atomic per column per block at the end.
// ---------------------------------------------------------------------------
__global__ void k_weighted_rowsum(const float* __restrict__ X,
                                  const unsigned int* __restrict__ counts,
                                  float* __restrict__ ls, int nrows) {
    int h = threadIdx.x;
    float acc = 0.0f;
    for (int n = blockIdx.x; n < nrows; n += gridDim.x) {
        float c = (float)counts[n];                 // scalar (uniform) load
        acc = fmaf(c, X[(size_t)n * H + h], acc);   // coalesced vector load
    }
    atomicAdd(&ls[h], acc);
}

// ---------------------------------------------------------------------------
// K3: v[i] = sum_j R[i][j] * ls[j].  Single block, 8 waves, one row per wave
// per iteration; lanes stride the row (coalesced 128B reads), then a
// __shfl_xor tree reduction (wave32).
// ---------------------------------------------------------------------------
__global__ void k_matvec_R(const float* __restrict__ R,
                           const float* __restrict__ ls,
                           float* __restrict__ v) {
    __shared__ float s[H];
    s[threadIdx.x] = ls[threadIdx.x];
    __syncthreads();
    int wave = threadIdx.x >> 5;
    int lane = threadIdx.x & 31;
    for (int i = wave; i < H; i += 8) {
        const float* row = R + (size_t)i * H;
        float acc = 0.0f;
        #pragma unroll
        for (int j = lane; j < H; j += 32) acc = fmaf(row[j], s[j], acc);
        #pragma unroll
        for (int off = 16; off > 0; off >>= 1) acc += __shfl_xor(acc, off, 32);
        if (lane == 0) v[i] = acc;
    }
}

// ---------------------------------------------------------------------------
// K4 (WMMA): t[n] = x_trg[n] . v  for all n, via V_WMMA_F32_16X16X4_F32.
//
//   D(16x16) += A(16x4) x B(4x16), accumulated over K = 0..256 step 4.
//   A: every row is the same v-chunk  -> A[m][k] = v[k0+k].
//      f32 16x4 A layout: lane l holds {A[M][kk], A[M][kk+1]} with
//      M = l&15, kk = k0 + (l>>4)*2  -> per-lane value is just v[kk],v[kk+1],
//      read as a broadcast ds_load_b64 from LDS.
//   B: B[k][n] = X[row0+n][k0+k] (a 4x16 tile of X^T). By the symmetric
//      row-striped layout, lane l loads the contiguous 8 bytes
//      X[row0+(l&15)][k0 + (l>>4)*2 .. +1].
//   With identical A rows, D[m][n] = sum_k v[k] X[row0+n][k] for every m,
//   so after the K loop lanes 0..15 hold t[row0+lane] in c[0].
//
//   EXEC must stay all-ones for WMMA: tail rows clamp their load address to
//   row 0 and only the final store is predicated.
// ---------------------------------------------------------------------------
__global__ void k_dot_rows_wmma(const float* __restrict__ X,
                                const float* __restrict__ v,
                                float* __restrict__ t, int nrows) {
    __shared__ float vs[H];
    vs[threadIdx.x] = v[threadIdx.x];
    __syncthreads();

    int wave = threadIdx.x >> 5;
    int lane = threadIdx.x & 31;
    int row0 = (blockIdx.x * 8 + wave) * 16;

    int r = row0 + (lane & 15);
    if (r >= nrows) r = 0;                       // clamp: keep EXEC full
    const float* xrow = X + (size_t)r * H;
    int khalf = (lane >> 4) * 2;                 // 0 for lanes 0-15, 2 for 16-31

    v8f c = {};
    #pragma unroll 8
    for (int k0 = 0; k0 < H; k0 += 4) {
        int kk = k0 + khalf;
        v2f a = *(const v2f*)(vs + kk);          // A fragment (v, replicated rows)
        v2f b = *(const v2f*)(xrow + kk);        // B fragment (X^T tile)
        c = __builtin_amdgcn_wmma_f32_16x16x4_f32(
                /*neg_a=*/false, a, /*neg_b=*/false, b,
                /*c_mod=*/(short)0, c, /*reuse_a=*/false, /*reuse_b=*/false);
    }

    int n = row0 + (lane & 15);
    if (lane < 16 && n < nrows) t[n] = c[0];     // D row 0: N = lane, VGPR 0
}

// ---------------------------------------------------------------------------
// K5: scores[e] = t[trg_idx[e]].  t is 200KB -> L2-resident gather.
// ---------------------------------------------------------------------------
__global__ void k_gather_scores(const float* __restrict__ t,
                                const int* __restrict__ trg_idx,
                                float* __restrict__ out, int E) {
    int i = blockIdx.x * blockDim.x + threadIdx.x;
    if (i < E) out[i] = t[trg_idx[i]];
}

// ---------------------------------------------------------------------------
extern "C" void kernel_launch(void* const* d_in, const int* in_sizes, int n_in,
                              void* d_out, int out_size, void* d_ws, size_t ws_size,
                              hipStream_t stream) {
    const float* x_src = (const float*)d_in[0];
    const float* x_trg = (const float*)d_in[1];
    const float* R     = (const float*)d_in[2];
    const int*   edges = (const int*)d_in[3];   // (2, E) flat: row0=src, row1=trg

    const int n_src = in_sizes[0] / H;
    const int n_trg = in_sizes[1] / H;
    const int E     = in_sizes[3] / 2;
    const int* src_idx = edges;
    const int* trg_idx = edges + E;
    float* out = (float*)d_out;

    // Workspace layout (256B-aligned slots): counts | left_sum | v | t
    char* ws = (char*)d_ws;
    size_t off = 0;
    unsigned int* counts = (unsigned int*)(ws + off); off += ((size_t)n_src * 4 + 255) & ~(size_t)255;
    float* ls            = (float*)(ws + off);        off += (H * 4 + 255) & ~(size_t)255;
    float* vvec          = (float*)(ws + off);        off += (H * 4 + 255) & ~(size_t)255;
    float* t             = (float*)(ws + off);

    // K0: zero counts + ls + v (contiguous-ish: do each region)
    {
        int nz = n_src;
        k_zero<<<(nz + 255) / 256, 256, 0, stream>>>(counts, nz);
        k_zero<<<1, 256, 0, stream>>>((unsigned int*)ls, H);
        k_zero<<<1, 256, 0, stream>>>((unsigned int*)vvec, H);
    }
    // K1: histogram of source indices
    k_hist<<<(E + 255) / 256, 256, 0, stream>>>(src_idx, counts, E);
    // K2: left_sum = sum_n counts[n] * x_src[n]
    k_weighted_rowsum<<<512, H, 0, stream>>>(x_src, counts, ls, n_src);
    // K3: v = R @ left_sum
    k_matvec_R<<<1, H, 0, stream>>>(R, ls, vvec);
    // K4: t[n] = x_trg[n] . v   (WMMA; 128 rows per 256-thread block)
    {
        int blocks = (n_trg + 127) / 128;
        k_dot_rows_wmma<<<blocks, 256, 0, stream>>>(x_trg, vvec, t, n_trg);
    }
    // K5: scores[e] = t[trg_idx[e]]
    k_gather_scores<<<(E + 255) / 256, 256, 0, stream>>>(t, trg_idx, out, E);
}